// HawkesBase_82016695485393
// MI455X (gfx1250) — compile-verified
//
#include <hip/hip_runtime.h>
#include <hip/hip_bf16.h>

typedef __attribute__((ext_vector_type(2))) float v2f;
typedef __attribute__((ext_vector_type(8))) float v8f;

#define CHUNK 400          // events per chunk (N = 200000 = 500 * 400)
#define TILE  16           // WMMA event-tile
#define NTYPES 64
#define NKER  4
#define NSTATE 256         // NTYPES * NKER

// ---------------------------------------------------------------------------
// Kernel 0: precompute W[j][n] = gamma[k]*alpha[k][m'][n]  (j = m'*4 + k),
//           asum[k*64+m] = sum_n alpha[k][m][n], zero the f64 accumulators.
// ---------------------------------------------------------------------------
__global__ __launch_bounds__(256) void hawkes_prep(const float* __restrict__ alpha,
                                                   const float* __restrict__ gamma,
                                                   float* __restrict__ W,
                                                   float* __restrict__ asum,
                                                   double* __restrict__ acc) {
  int b = blockIdx.x, tid = threadIdx.x;
  if (b < 64) {
    int idx = b * 256 + tid;           // 0..16383 == j*64 + n
    int j = idx >> 6, n = idx & 63;
    int m = j >> 2, k = j & 3;
    W[idx] = gamma[k] * alpha[(k * NTYPES + m) * NTYPES + n];
  } else {
    int k = tid >> 6, m = tid & 63;    // tid == k*64+m
    float s = 0.f;
#pragma unroll 8
    for (int n = 0; n < NTYPES; ++n) s += alpha[(k * NTYPES + m) * NTYPES + n];
    asum[tid] = s;
    if (tid < 2) acc[tid] = 0.0;
  }
}

// ---------------------------------------------------------------------------
// Kernel 1: per-chunk scan summary. Thread tid owns state element
// j = tid (m' = tid>>2, k = tid&3). Chunk transform: g_out = A*g_in + B.
// ---------------------------------------------------------------------------
__global__ __launch_bounds__(256) void hawkes_scan_partial(const float* __restrict__ t,
                                                           const int* __restrict__ mi,
                                                           const float* __restrict__ gamma,
                                                           float* __restrict__ Aws,
                                                           float* __restrict__ Bws) {
  __shared__ float tsh[CHUNK + 1];
  __shared__ int   msh[CHUNK];
  int p = blockIdx.x, tid = threadIdx.x;
  int s = p * CHUNK;
  for (int q = tid; q <= CHUNK; q += 256) {
    int gi = s - 1 + q;
    tsh[q] = t[gi < 0 ? 0 : gi];                 // tsh[0] = t[s-1] (t[0] for p==0 -> dt0 = 0)
  }
  for (int q = tid; q < CHUNK; q += 256) {
    int gi = s - 1 + q;
    msh[q] = (gi < 0) ? -1 : mi[gi];             // m_{i-1}; -1 sentinel => zero one-hot
  }
  __syncthreads();

  int mprime = tid >> 2, k = tid & 3;
  float gk = gamma[k];
  float a = 1.f, b = 0.f;
  for (int q = 0; q < CHUNK; ++q) {
    float dt = tsh[q + 1] - tsh[q];
    float d = __expf(-gk * dt);
    a *= d;
    b = d * (b + ((msh[q] == mprime) ? 1.f : 0.f));
  }
  Bws[p * NSTATE + tid] = b;
  if (tid < NKER) Aws[p * NKER + tid] = a;       // tid<4 => k == tid
}

// ---------------------------------------------------------------------------
// Kernel 2: sequential combine of chunk transforms (jax combine:
// (a1,b1)∘(a2,b2) = (a1*a2, a2*b1+b2)). Gstart[p] = state before chunk p.
// ---------------------------------------------------------------------------
__global__ __launch_bounds__(256) void hawkes_carry(const float* __restrict__ Aws,
                                                    const float* __restrict__ Bws,
                                                    float* __restrict__ Gstart, int P) {
  int j = threadIdx.x, k = j & 3;
  float cur = 0.f;                               // g[-1] = 0
  for (int p = 0; p < P; ++p) {
    Gstart[p * NSTATE + j] = cur;
    cur = Aws[p * NKER + k] * cur + Bws[p * NSTATE + j];
  }
}

// ---------------------------------------------------------------------------
// Kernel 3: replay each chunk. Per 16-event tile:
//   (a) 256-thread VALU scan -> g-tile in LDS (16 x 256 f32)
//   (b) ALL 8 waves run WMMA: wave w and w+4 split the 256-deep contraction
//       of output column tile (w&3); 32 x v_wmma_f32_16x16x4_f32 per wave,
//       B-operand held entirely in VGPRs (hoisted, invariant per block).
//   (c) threads 0..15: gather lambda (summing both halves), logf,
//       expm1f survival, f64 accumulate
// ---------------------------------------------------------------------------
__global__ __launch_bounds__(256) void hawkes_main(const float* __restrict__ t,
                                                   const int* __restrict__ mi,
                                                   const float* __restrict__ mu,
                                                   const float* __restrict__ gamma,
                                                   const float* __restrict__ W,
                                                   const float* __restrict__ asum,
                                                   const float* __restrict__ Gstart,
                                                   const int* __restrict__ Traw,
                                                   double* __restrict__ acc) {
  __shared__ float tsh[CHUNK + 1];
  __shared__ int   msh[CHUNK + 1];
  __shared__ float gt[TILE][NSTATE + 1];         // +1 pad: stride 257 -> conflict-free columns
  __shared__ float lamsh[2][TILE][NTYPES];       // two contraction halves
  __shared__ double rlog[TILE], rcomp[TILE];

  int p = blockIdx.x, tid = threadIdx.x;
  int s = p * CHUNK;

  int iv = Traw[0];
  float Tval = (iv > 0 && iv < (1 << 20)) ? (float)iv : __int_as_float(iv);

  for (int q = tid; q <= CHUNK; q += 256) {
    int gi = s - 1 + q;
    tsh[q] = t[gi < 0 ? 0 : gi];
    msh[q] = (gi < 0) ? -1 : mi[gi];
  }
  __syncthreads();

  int mprime = tid >> 2, k = tid & 3;
  float gk = gamma[k];
  float g = Gstart[p * NSTATE + tid];            // g[s-1]
  int wave = tid >> 5, lane = tid & 31;
  float gm0 = gamma[0], gm1 = gamma[1], gm2 = gamma[2], gm3 = gamma[3];
  double alog = 0.0, acomp = 0.0;

  // WMMA geometry: wave w handles column tile (w&3), contraction half (w>>2)
  int col0   = (wave & 3) * 16;                  // output N tile base
  int half   = wave >> 2;                        // 0: j in [0,128), 1: j in [128,256)
  int chbase = half * 32;                        // contraction slices (of depth 4)
  int row    = lane & 15;                        // A: M index (event)
  int kofs   = (lane < 16) ? 0 : 2;              // A/B: K sub-offset per ISA layout
  int nloc   = lane & 15;                        // B: N index
  int mb     = (lane < 16) ? 0 : 8;              // D layout: VGPR r -> M=r / M=r+8

  // Hoist this wave's B-operand slice of W into registers (64 VGPRs),
  // invariant across all 25 tiles of the chunk.
  v2f breg[32];
#pragma unroll
  for (int c = 0; c < 32; ++c) {
    int j0 = (chbase + c) * 4;
    breg[c][0] = W[(j0 + kofs) * NTYPES + col0 + nloc];
    breg[c][1] = W[(j0 + kofs + 1) * NTYPES + col0 + nloc];
  }

  for (int tile = 0; tile < CHUNK / TILE; ++tile) {
    int base = tile * TILE;
    // (a) state scan phase: 16 sequential steps, all 256 threads
#pragma unroll
    for (int q = 0; q < TILE; ++q) {
      float dt = tsh[base + q + 1] - tsh[base + q];
      float d = __expf(-gk * dt);
      g = d * (g + ((msh[base + q] == mprime) ? 1.f : 0.f));
      gt[q][tid] = g;
    }
    __syncthreads();

    // (b) WMMA phase: every wave, full EXEC, 32 x wmma f32 16x16x4
    {
      v8f accv = {};
#pragma unroll
      for (int c = 0; c < 32; ++c) {
        int j0 = (chbase + c) * 4;
        v2f av;
        av[0] = gt[row][j0 + kofs];
        av[1] = gt[row][j0 + kofs + 1];
        accv = __builtin_amdgcn_wmma_f32_16x16x4_f32(
            /*neg_a=*/false, av, /*neg_b=*/false, breg[c],
            /*c_mod=*/(short)0, accv, /*reuse_a=*/false, /*reuse_b=*/false);
      }
#pragma unroll
      for (int r = 0; r < 8; ++r) lamsh[half][mb + r][col0 + nloc] = accv[r];
    }
    __syncthreads();

    // (c) gather + accumulate: thread q handles event s+base+q
    if (tid < TILE) {
      int q = tid;
      int m  = msh[base + q + 1];                // m_i
      float ti = tsh[base + q + 1];              // t_i
      float lam = mu[m] + lamsh[0][q][m] + lamsh[1][q][m];
      alog += (double)__logf(lam);
      float rem = Tval - ti;
      float sv = asum[m]            * (-expm1f(-gm0 * rem))
               + asum[NTYPES + m]   * (-expm1f(-gm1 * rem))
               + asum[2*NTYPES + m] * (-expm1f(-gm2 * rem))
               + asum[3*NTYPES + m] * (-expm1f(-gm3 * rem));
      acomp += (double)sv;
    }
    // next iteration's __syncthreads (after scan phase) orders gather vs. lamsh reuse
  }

  if (tid < TILE) { rlog[tid] = alog; rcomp[tid] = acomp; }
  __syncthreads();
  if (tid == 0) {
    double sl = 0.0, sc = 0.0;
#pragma unroll
    for (int q = 0; q < TILE; ++q) { sl += rlog[q]; sc += rcomp[q]; }
    atomicAdd(&acc[0], sl);                      // global_atomic_add_f64
    atomicAdd(&acc[1], sc);
  }
}

// ---------------------------------------------------------------------------
// Kernel 4: final scalar assembly
// ---------------------------------------------------------------------------
__global__ void hawkes_final(const float* __restrict__ mu,
                             const double* __restrict__ acc,
                             const int* __restrict__ Traw,
                             int N, float* __restrict__ out) {
  if (threadIdx.x == 0 && blockIdx.x == 0) {
    double musum = 0.0;
    for (int m = 0; m < NTYPES; ++m) musum += (double)mu[m];
    int iv = Traw[0];
    float Tval = (iv > 0 && iv < (1 << 20)) ? (float)iv : __int_as_float(iv);
    double comp = (double)Tval * musum + acc[1];
    out[0] = (float)((comp - acc[0]) / (double)N);
  }
}

// ---------------------------------------------------------------------------
extern "C" void kernel_launch(void* const* d_in, const int* in_sizes, int n_in,
                              void* d_out, int out_size, void* d_ws, size_t ws_size,
                              hipStream_t stream) {
  const float* mu    = (const float*)d_in[0];   // (64,)
  const float* alpha = (const float*)d_in[1];   // (4,64,64)
  const float* gamma = (const float*)d_in[2];   // (4,)
  const float* t     = (const float*)d_in[3];   // (200000,)
  const int*   mi    = (const int*)d_in[4];     // (200000,)
  const int*   Traw  = (const int*)d_in[5];     // scalar T

  int N = in_sizes[3];
  int P = N / CHUNK;                            // 500 chunks

  // Workspace layout (~1.1 MB total):
  char* ws = (char*)d_ws;
  float*  W    = (float*)ws;                                  // 16384 f32 (64 KB)
  float*  asum = (float*)(ws + 65536);                        //   256 f32
  double* acc  = (double*)(ws + 66560);                       //     2 f64
  char* base   = ws + 66576;
  float* Aws = (float*)base;                                  // P*4 f32
  float* Bws = (float*)(base + (size_t)P * NKER * 4);         // P*256 f32
  float* Gst = (float*)(base + (size_t)P * NKER * 4 + (size_t)P * NSTATE * 4);

  hawkes_prep<<<65, 256, 0, stream>>>(alpha, gamma, W, asum, acc);
  hawkes_scan_partial<<<P, 256, 0, stream>>>(t, mi, gamma, Aws, Bws);
  hawkes_carry<<<1, 256, 0, stream>>>(Aws, Bws, Gst, P);
  hawkes_main<<<P, 256, 0, stream>>>(t, mi, mu, gamma, W, asum, Gst, Traw, acc);
  hawkes_final<<<1, 64, 0, stream>>>(mu, acc, Traw, N, (float*)d_out);
}